// IDAM_50122268344796
// MI455X (gfx1250) — compile-verified
//
#include <hip/hip_runtime.h>
#include <hip/hip_bf16.h>
#include <math.h>

// ---------------------------------------------------------------------------
// IDAM forward for MI455X (gfx1250, wave32). All channel-matmuls lowered to
// V_WMMA_F32_16X16X4_F32. BatchNorm (training mode) handled as:
//   conv (+stats via LDS tile row-sums) -> finalize(scale,shift) -> next conv
//   applies scale/shift+ReLU while loading its B-fragments.
// The 132-ch pairwise feature tensor is staged per-block in LDS (never hits
// DRAM); only simfeat (B*64*NP*NP) is materialized per iteration.
// ---------------------------------------------------------------------------

#define B_   8
#define N_   2048
#define NP_  204
#define NPP_ (NP_ * NP_)          // 41616
#define NPBLK ((NPP_ + 63) / 64)  // 651 position-tiles of 64
#define PITCH 68                  // LDS row pitch (floats), 16B-aligned rows

typedef __attribute__((ext_vector_type(2))) float v2f;
typedef __attribute__((ext_vector_type(8))) float v8f;

#define WMMA_F32(ACC, A, Bf) \
  (ACC) = __builtin_amdgcn_wmma_f32_16x16x4_f32(false, (A), false, (Bf), (short)0, (ACC), false, false)

// ---------------- workspace layout (float units) ----------------
static constexpr size_t O_LOSS   = 0;                                   // 8 accumulators
static constexpr size_t O_SUM0   = 8;
static constexpr size_t O_SSQ0   = O_SUM0 + 128;
static constexpr size_t O_SCL0   = O_SSQ0 + 128;
static constexpr size_t O_SHF0   = O_SCL0 + 128;
static constexpr size_t O_SUM1   = O_SHF0 + 128;
static constexpr size_t O_SSQ1   = O_SUM1 + 128;
static constexpr size_t O_SCL1   = O_SSQ1 + 128;
static constexpr size_t O_SHF1   = O_SCL1 + 128;
static constexpr size_t O_MINIDX = O_SHF1 + 128;                        // B*N ints
static constexpr size_t O_MATCHF = O_MINIDX + (size_t)B_ * N_;
static constexpr size_t O_TEMB   = O_MATCHF + (size_t)B_ * N_;          // B*64*N
static constexpr size_t O_SEMB   = O_TEMB + (size_t)B_ * 64 * N_;
static constexpr size_t O_HA     = O_SEMB + (size_t)B_ * 64 * N_;       // B*128*N
static constexpr size_t O_HB     = O_HA + (size_t)B_ * 128 * N_;        // B*64*N
static constexpr size_t O_SSIGF  = O_HB + (size_t)B_ * 64 * N_;
static constexpr size_t O_TSIGF  = O_SSIGF + (size_t)B_ * N_;
static constexpr size_t O_MATCHG = O_TSIGF + (size_t)B_ * N_;           // B*NP
static constexpr size_t O_SRCP   = O_MATCHG + (size_t)B_ * NP_;
static constexpr size_t O_TGTP   = O_SRCP + (size_t)B_ * 3 * NP_;
static constexpr size_t O_SEMBG  = O_TGTP + (size_t)B_ * 3 * NP_;
static constexpr size_t O_TEMBG  = O_SEMBG + (size_t)B_ * 64 * NP_;
static constexpr size_t O_SSIGG  = O_TEMBG + (size_t)B_ * 64 * NP_;
static constexpr size_t O_TSIGG  = O_SSIGG + (size_t)B_ * NP_;
static constexpr size_t O_SRCCUR = O_TSIGG + (size_t)B_ * NP_;
static constexpr size_t O_W64    = O_SRCCUR + (size_t)B_ * 3 * NP_;     // B*64*NP
static constexpr size_t O_WHS    = O_W64 + (size_t)B_ * 64 * NP_;       // B*32*NP
static constexpr size_t O_WSIG   = O_WHS + (size_t)B_ * 32 * NP_;
static constexpr size_t O_WNORM  = O_WSIG + (size_t)B_ * NP_;
static constexpr size_t O_CORR   = O_WNORM + (size_t)B_ * NP_;          // B*NP ints
static constexpr size_t O_RCUR   = O_CORR + (size_t)B_ * NP_;
static constexpr size_t O_TCUR   = O_RCUR + 72;
static constexpr size_t O_SIMFIN = O_TCUR + 24;                         // B*NPP
static constexpr size_t O_SIMFEA = O_SIMFIN + (size_t)B_ * NPP_;        // B*64*NPP

// ---------------- input index map (jax pytree / sorted dict keys) ----------
enum {
  IN_SRC = 0, IN_TGT, IN_RGT, IN_TGT_T, IN_SRCIDX,
  IN_EMB0_B = 5, IN_EMB0_G, IN_EMB0_W, IN_EMB1_B, IN_EMB1_G, IN_EMB1_W,
  IN_EMB2_BIAS, IN_EMB2_W,
  IN_SIG0_B = 13, IN_SIG0_G, IN_SIG0_W, IN_SIG1_B, IN_SIG1_G, IN_SIG1_W,
  IN_SIG2_BIAS, IN_SIG2_W,
  IN_SIM1 = 21,  // per iter i: +5i -> {b, g, w, bias, w2}
  IN_SIM2 = 36,
  IN_WFC  = 51,
};

// ============================ utility kernels ==============================
__global__ void fill_kernel(float* p, float v, int n) {
  int t = blockIdx.x * 256 + threadIdx.x;
  if (t < n) p[t] = v;
}

__global__ void init_Rt_kernel(float* Rcur, float* tcur) {
  int t = threadIdx.x;
  if (t < B_ * 9) { int r = t % 9; Rcur[t] = (r == 0 || r == 4 || r == 8) ? 1.f : 0.f; }
  if (t < B_ * 3) tcur[t] = 0.f;
}

__global__ void bn_finalize_kernel(const float* __restrict__ sum, const float* __restrict__ ssq,
                                   const float* __restrict__ g, const float* __restrict__ bb,
                                   float* scl, float* shf, int nch, float count) {
  int c = threadIdx.x;
  if (c < nch) {
    float m = sum[c] / count;
    float v = ssq[c] / count - m * m;
    float s = g[c] * rsqrtf(fmaxf(v, 0.f) + 1e-5f);
    scl[c] = s;
    shf[c] = bb[c] - m * s;
  }
}

// row-sum helper over an LDS slab row (len multiple of 4, 16B-aligned rows)
__device__ __forceinline__ void lds_row_stats(const float* row, int len, float* s_out, float* q_out) {
  float s = 0.f, q = 0.f;
  const float4* r4 = (const float4*)row;
  for (int c = 0; c < (len >> 2); ++c) {
    float4 x = r4[c];
    s += x.x + x.y + x.z + x.w;
    q += x.x * x.x + x.y * x.y + x.z * x.z + x.w * x.w;
  }
  *s_out = s;
  *q_out = q;
}

// ===================== GT correspondence (argmin over N) ===================
__global__ void corr_kernel(const float* __restrict__ src, const float* __restrict__ tgt,
                            const float* __restrict__ Rgt, const float* __restrict__ tgt_t,
                            int* __restrict__ min_idx, float* __restrict__ match_full) {
  __shared__ float tx[3][256];
  int b = blockIdx.y;
  int n = blockIdx.x * 256 + threadIdx.x;
  float sg[3];
#pragma unroll
  for (int i = 0; i < 3; ++i) {
    float s = tgt_t[b * 3 + i];
#pragma unroll
    for (int j = 0; j < 3; ++j) s += Rgt[(b * 3 + i) * 3 + j] * src[((size_t)b * 3 + j) * N_ + n];
    sg[i] = s;
  }
  float best = 3.4e38f;
  int bi = 0;
  for (int m0 = 0; m0 < N_; m0 += 256) {
    __syncthreads();
#pragma unroll
    for (int c = 0; c < 3; ++c) tx[c][threadIdx.x] = tgt[((size_t)b * 3 + c) * N_ + m0 + threadIdx.x];
    __syncthreads();
    for (int mm = 0; mm < 256; ++mm) {
      float d0 = sg[0] - tx[0][mm], d1 = sg[1] - tx[1][mm], d2 = sg[2] - tx[2][mm];
      float d = d0 * d0 + d1 * d1 + d2 * d2;
      if (d < best) { best = d; bi = m0 + mm; }
    }
  }
  min_idx[b * N_ + n] = bi;
  match_full[b * N_ + n] = (sqrtf(fmaxf(best, 0.f)) < 0.05f) ? 1.f : 0.f;
}

// ============= generic WMMA conv (ksize=1 channel matmul) + BN stats =======
// out[b,o,n] = sum_c W[o,c] * f(x[b,c,n]);  f = optional scale/shift + ReLU.
// Stats: waves deposit masked D-tiles into a [16][132] LDS slab; 16 threads
// row-sum (ds_load_b128) and issue 2 global atomics per channel per block.
template <int CIN_PAD, int CIN, int COUT, bool IN_BN, bool ADD_BIAS, bool STATS>
__global__ void convbn_kernel(const float* __restrict__ x, const float* __restrict__ Wt,
                              const float* __restrict__ bias,
                              const float* __restrict__ scl, const float* __restrict__ shf,
                              float* __restrict__ out,
                              float* __restrict__ gsum, float* __restrict__ gssq, int P) {
  __shared__ float red[STATS ? 16 * 132 : 1];
  int tid = threadIdx.x;
  int wave = tid >> 5, lane = tid & 31, half = lane >> 4, l16 = lane & 15;
  int b = blockIdx.z;
  int m0 = blockIdx.y * 16;
  int n0 = (blockIdx.x * 8 + wave) * 16;
  const float* xb = x + (size_t)b * CIN * P;
  int m = m0 + l16;
  int n = n0 + l16;
  bool nv = n < P;
  int nc = nv ? n : (P - 1);  // clamped column: loads always valid
  v8f acc = {0.f, 0.f, 0.f, 0.f, 0.f, 0.f, 0.f, 0.f};
#pragma unroll 4
  for (int k = 0; k < CIN_PAD; k += 4) {
    int ka = k + half * 2;
    v2f a, bf;
    float w0, w1, b0, b1;
    if (CIN_PAD != CIN) {  // compile-time: only the 3->64 layer takes this path
      w0 = (ka < CIN) ? Wt[m * CIN + ka] : 0.f;
      w1 = (ka + 1 < CIN) ? Wt[m * CIN + ka + 1] : 0.f;
      b0 = (ka < CIN) ? xb[(size_t)ka * P + nc] : 0.f;
      b1 = (ka + 1 < CIN) ? xb[(size_t)(ka + 1) * P + nc] : 0.f;
    } else {
      w0 = Wt[m * CIN + ka];
      w1 = Wt[m * CIN + ka + 1];
      b0 = xb[(size_t)ka * P + nc];
      b1 = xb[(size_t)(ka + 1) * P + nc];
    }
    if (IN_BN) {
      b0 = fmaxf(b0 * scl[ka] + shf[ka], 0.f);
      b1 = fmaxf(b1 * scl[ka + 1] + shf[ka + 1], 0.f);
    }
    a.x = w0; a.y = w1;
    bf.x = b0; bf.y = b1;
    WMMA_F32(acc, a, bf);
  }
  float* ob = out + (size_t)b * COUT * P;
#pragma unroll
  for (int v = 0; v < 8; ++v) {
    int ch = m0 + v + half * 8;
    float val = acc[v];
    if (ADD_BIAS) val += bias[ch];
    if (nv) ob[(size_t)ch * P + n] = val;
    if (STATS) red[(v + half * 8) * 132 + wave * 16 + l16] = nv ? val : 0.f;
  }
  if (STATS) {
    __syncthreads();
    if (tid < 16) {
      float s, q;
      lds_row_stats(&red[tid * 132], 128, &s, &q);
      atomicAdd(&gsum[m0 + tid], s);
      atomicAdd(&gssq[m0 + tid], q);
    }
  }
}

// last conv layer with Cout=1 (sig head), VALU
__global__ void conv_last1_kernel(const float* __restrict__ hin, const float* __restrict__ scl,
                                  const float* __restrict__ shf, const float* __restrict__ Wt,
                                  const float* __restrict__ bias, float* __restrict__ out, int P) {
  int t = blockIdx.x * 256 + threadIdx.x;
  if (t >= B_ * P) return;
  int b = t / P, n = t % P;
  float s = bias[0];
#pragma unroll
  for (int c = 0; c < 32; ++c) {
    float h = hin[((size_t)b * 32 + c) * (size_t)P + n];
    s += Wt[c] * fmaxf(h * scl[c] + shf[c], 0.f);
  }
  out[t] = s;
}

// ========================= point pruning gathers ===========================
__global__ void gather_kernel(const float* __restrict__ src, const float* __restrict__ tgt,
                              const int* __restrict__ src_idx, const int* __restrict__ min_idx,
                              const float* __restrict__ match_full,
                              const float* __restrict__ semb_f, const float* __restrict__ temb_f,
                              const float* __restrict__ ssig_f, const float* __restrict__ tsig_f,
                              float* srcp, float* tgtp, float* sembg, float* tembg,
                              float* ssigg, float* tsigg, float* matchg, float* srccur,
                              float* lossacc) {
  int t = blockIdx.x * 256 + threadIdx.x;
  if (t >= B_ * NP_) return;
  int b = t / NP_, k = t % NP_;
  int si = src_idx[t];
  int ti = min_idx[b * N_ + si];
  float mg = match_full[b * N_ + si];
  matchg[t] = mg;
  atomicAdd(&lossacc[1], mg);
#pragma unroll
  for (int c = 0; c < 3; ++c) {
    float sv = src[((size_t)b * 3 + c) * N_ + si];
    srcp[((size_t)b * 3 + c) * NP_ + k] = sv;
    srccur[((size_t)b * 3 + c) * NP_ + k] = sv;
    tgtp[((size_t)b * 3 + c) * NP_ + k] = tgt[((size_t)b * 3 + c) * N_ + ti];
  }
  for (int c = 0; c < 64; ++c) {
    sembg[((size_t)b * 64 + c) * NP_ + k] = semb_f[((size_t)b * 64 + c) * N_ + si];
    tembg[((size_t)b * 64 + c) * NP_ + k] = temb_f[((size_t)b * 64 + c) * N_ + ti];
  }
  ssigg[t] = ssig_f[b * N_ + si];
  tsigg[t] = tsig_f[b * N_ + ti];
}

// ============ pairwise feature tile staged in LDS (132 x 64) ===============
__device__ __forceinline__ void build_feat_tile(float (*feat)[PITCH], int* iofs, int* jofs,
                                                int b, int p0, int tid,
                                                const float* __restrict__ se,
                                                const float* __restrict__ te,
                                                const float* __restrict__ scur,
                                                const float* __restrict__ tp) {
  if (tid < 64) {
    int p = p0 + tid;
    if (p > NPP_ - 1) p = NPP_ - 1;  // clamp: padded cols replicate last position
    int i = p / NP_, j = p - i * NP_;
    iofs[tid] = i;
    jofs[tid] = j;
    float dx = scur[((size_t)b * 3 + 0) * NP_ + i] - tp[((size_t)b * 3 + 0) * NP_ + j];
    float dy = scur[((size_t)b * 3 + 1) * NP_ + i] - tp[((size_t)b * 3 + 1) * NP_ + j];
    float dz = scur[((size_t)b * 3 + 2) * NP_ + i] - tp[((size_t)b * 3 + 2) * NP_ + j];
    float d = sqrtf(dx * dx + dy * dy + dz * dz);
    float inv = 1.f / (d + 1e-8f);
    feat[128][tid] = d;
    feat[129][tid] = dx * inv;
    feat[130][tid] = dy * inv;
    feat[131][tid] = dz * inv;
  }
  __syncthreads();
  for (int idx = tid; idx < 128 * 64; idx += 256) {
    int c = idx >> 6, pp = idx & 63;
    feat[c][pp] = (c < 64) ? se[((size_t)b * 64 + c) * NP_ + iofs[pp]]
                           : te[((size_t)b * 64 + (c - 64)) * NP_ + jofs[pp]];
  }
  __syncthreads();
}

// Pass 1: h1 = W1a @ feat, stats only (tiles land in LDS, row-summed)
__global__ void sim_h1_stats_kernel(const float* __restrict__ W1a,
                                    const float* __restrict__ se, const float* __restrict__ te,
                                    const float* __restrict__ scur, const float* __restrict__ tp,
                                    float* __restrict__ gsum, float* __restrict__ gssq) {
  __shared__ float feat[132][PITCH];
  __shared__ float h1s[64][PITCH];
  __shared__ int iofs[64], jofs[64];
  int tid = threadIdx.x;
  int b = blockIdx.y;
  int p0 = blockIdx.x * 64;
  build_feat_tile(feat, iofs, jofs, b, p0, tid, se, te, scur, tp);
  int wave = tid >> 5, lane = tid & 31, half = lane >> 4, l16 = lane & 15;
  for (int t = wave; t < 16; t += 8) {
    int mt = t >> 2, nt = t & 3;
    int m = mt * 16 + l16;
    int col = nt * 16 + l16;
    bool pv = (p0 + col) < NPP_;
    v8f acc = {0.f, 0.f, 0.f, 0.f, 0.f, 0.f, 0.f, 0.f};
#pragma unroll 3
    for (int k = 0; k < 132; k += 4) {
      int ka = k + half * 2;
      v2f a, bf;
      a.x = W1a[m * 132 + ka];
      a.y = W1a[m * 132 + ka + 1];
      bf.x = feat[ka][col];
      bf.y = feat[ka + 1][col];
      WMMA_F32(acc, a, bf);
    }
#pragma unroll
    for (int v = 0; v < 8; ++v) {
      int ch = mt * 16 + v + half * 8;
      h1s[ch][col] = pv ? acc[v] : 0.f;
    }
  }
  __syncthreads();
  if (tid < 64) {
    float s, q;
    lds_row_stats(&h1s[tid][0], 64, &s, &q);
    atomicAdd(&gsum[tid], s);
    atomicAdd(&gssq[tid], q);
  }
}

// Pass 2: recompute h1 (from LDS feat), BN+ReLU -> LDS, simfeat = W1b@h1r+b1b
// (store to DRAM + LDS), then fused h2 = W2a@simfeat stats (tiles -> LDS).
__global__ void sim_feat_kernel(const float* __restrict__ W1a, const float* __restrict__ sclA,
                                const float* __restrict__ shfA, const float* __restrict__ W1b,
                                const float* __restrict__ b1b, const float* __restrict__ W2a,
                                const float* __restrict__ se, const float* __restrict__ te,
                                const float* __restrict__ scur, const float* __restrict__ tp,
                                float* __restrict__ simfeat,
                                float* __restrict__ gsum, float* __restrict__ gssq) {
  __shared__ float smem[(132 + 64) * PITCH];
  float(*feat)[PITCH] = (float(*)[PITCH])smem;                 // stages 0-1
  float(*h1s)[PITCH] = (float(*)[PITCH])(smem + 132 * PITCH);  // stage 1 out, stage 3 h2 slab
  float(*sfs)[PITCH] = (float(*)[PITCH])smem;                  // stage 2 out (reuses feat)
  __shared__ int iofs[64], jofs[64];
  int tid = threadIdx.x;
  int b = blockIdx.y;
  int p0 = blockIdx.x * 64;
  build_feat_tile(feat, iofs, jofs, b, p0, tid, se, te, scur, tp);
  int wave = tid >> 5, lane = tid & 31, half = lane >> 4, l16 = lane & 15;
  // stage 1: h1 tiles -> BN+ReLU -> LDS
  for (int t = wave; t < 16; t += 8) {
    int mt = t >> 2, nt = t & 3;
    int m = mt * 16 + l16;
    int col = nt * 16 + l16;
    bool pv = (p0 + col) < NPP_;
    v8f acc = {0.f, 0.f, 0.f, 0.f, 0.f, 0.f, 0.f, 0.f};
#pragma unroll 3
    for (int k = 0; k < 132; k += 4) {
      int ka = k + half * 2;
      v2f a, bf;
      a.x = W1a[m * 132 + ka];
      a.y = W1a[m * 132 + ka + 1];
      bf.x = feat[ka][col];
      bf.y = feat[ka + 1][col];
      WMMA_F32(acc, a, bf);
    }
#pragma unroll
    for (int v = 0; v < 8; ++v) {
      int ch = mt * 16 + v + half * 8;
      h1s[ch][col] = pv ? fmaxf(acc[v] * sclA[ch] + shfA[ch], 0.f) : 0.f;
    }
  }
  __syncthreads();
  // stage 2: simfeat = W1b @ h1r + bias -> global + LDS (sfs aliases feat)
  for (int t = wave; t < 16; t += 8) {
    int mt = t >> 2, nt = t & 3;
    int m = mt * 16 + l16;
    int col = nt * 16 + l16;
    v8f acc = {0.f, 0.f, 0.f, 0.f, 0.f, 0.f, 0.f, 0.f};
#pragma unroll
    for (int k = 0; k < 64; k += 4) {
      int ka = k + half * 2;
      v2f a, bf;
      a.x = W1b[m * 64 + ka];
      a.y = W1b[m * 64 + ka + 1];
      bf.x = h1s[ka][col];
      bf.y = h1s[ka + 1][col];
      WMMA_F32(acc, a, bf);
    }
#pragma unroll
    for (int v = 0; v < 8; ++v) {
      int ch = mt * 16 + v + half * 8;
      float val = acc[v] + b1b[ch];
      sfs[ch][col] = val;
      size_t p = (size_t)p0 + col;
      if (p < NPP_) simfeat[((size_t)(b * 64 + ch)) * NPP_ + p] = val;
    }
  }
  __syncthreads();
  // stage 3: h2 = W2a @ simfeat; raw masked tiles into h1s slab for stats
  {
    int mt = wave >> 2, nt = wave & 3;
    int m = mt * 16 + l16;
    int col = nt * 16 + l16;
    v8f acc = {0.f, 0.f, 0.f, 0.f, 0.f, 0.f, 0.f, 0.f};
#pragma unroll
    for (int k = 0; k < 64; k += 4) {
      int ka = k + half * 2;
      v2f a, bf;
      a.x = W2a[m * 64 + ka];
      a.y = W2a[m * 64 + ka + 1];
      bf.x = sfs[ka][col];
      bf.y = sfs[ka + 1][col];
      WMMA_F32(acc, a, bf);
    }
    bool pv = (p0 + col) < NPP_;
#pragma unroll
    for (int v = 0; v < 8; ++v) {
      int ch = mt * 16 + v + half * 8;  // < 32
      h1s[ch][col] = pv ? acc[v] : 0.f;
    }
  }
  __syncthreads();
  if (tid < 32) {
    float s, q;
    lds_row_stats(&h1s[tid][0], 64, &s, &q);
    atomicAdd(&gsum[tid], s);
    atomicAdd(&gssq[tid], q);
  }
}

// weights64[b,c,i] = max_j simfeat[b,c,i,j]  (one wave per row)
__global__ void rowmax_kernel(const float* __restrict__ simfeat, float* __restrict__ w64) {
  int r = blockIdx.x * 8 + (threadIdx.x >> 5);
  int lane = threadIdx.x & 31;
  int i = r % NP_;
  int bc = r / NP_;
  const float* row = simfeat + (size_t)bc * NPP_ + (size_t)i * NP_;
  float m = -3.4e38f;
  for (int j = lane; j < NP_; j += 32) m = fmaxf(m, row[j]);
  for (int o = 16; o; o >>= 1) m = fmaxf(m, __shfl_xor(m, o, 32));
  if (!lane) w64[(size_t)bc * NP_ + i] = m;
}

// Pass 3: h2 = W2a@simfeat (BN+ReLU), sim = W2b@h2r + bias, clip +-20
__global__ void sim_final_kernel(const float* __restrict__ simfeat, const float* __restrict__ W2a,
                                 const float* __restrict__ scl2, const float* __restrict__ shf2,
                                 const float* __restrict__ W2b, const float* __restrict__ b2b,
                                 float* __restrict__ simfinal) {
  __shared__ float sfs[64][PITCH];
  __shared__ float h2s[32][PITCH];
  int tid = threadIdx.x;
  int wave = tid >> 5, lane = tid & 31, half = lane >> 4, l16 = lane & 15;
  int b = blockIdx.y;
  size_t p0 = (size_t)blockIdx.x * 64;
  for (int idx = tid; idx < 64 * 64; idx += 256) {
    int ch = idx >> 6, pp = idx & 63;
    size_t p = p0 + pp;
    if (p > NPP_ - 1) p = NPP_ - 1;
    sfs[ch][pp] = simfeat[((size_t)(b * 64 + ch)) * NPP_ + p];
  }
  __syncthreads();
  {
    int mt = wave >> 2, nt = wave & 3;
    int m = mt * 16 + l16;
    int col = nt * 16 + l16;
    v8f acc = {0.f, 0.f, 0.f, 0.f, 0.f, 0.f, 0.f, 0.f};
#pragma unroll
    for (int k = 0; k < 64; k += 4) {
      int ka = k + half * 2;
      v2f a, bf;
      a.x = W2a[m * 64 + ka];
      a.y = W2a[m * 64 + ka + 1];
      bf.x = sfs[ka][col];
      bf.y = sfs[ka + 1][col];
      WMMA_F32(acc, a, bf);
    }
#pragma unroll
    for (int v = 0; v < 8; ++v) {
      int ch = mt * 16 + v + half * 8;
      h2s[ch][col] = fmaxf(acc[v] * scl2[ch] + shf2[ch], 0.f);
    }
  }
  __syncthreads();
  if (tid < 64) {
    size_t p = p0 + tid;
    if (p < NPP_) {
      float s = b2b[0];
#pragma unroll
      for (int c = 0; c < 32; ++c) s += W2b[c] * h2s[c][tid];
      s = fminf(fmaxf(s, -20.f), 20.f);
      simfinal[(size_t)b * NPP_ + p] = s;
    }
  }
}

// ================== row/col softmax stats, argmax, losses ==================
__global__ void row_softmax_kernel(const float* __restrict__ simfinal,
                                   const float* __restrict__ matchg,
                                   const float* __restrict__ ssig, int* __restrict__ corr,
                                   float* __restrict__ lossacc, int isFirst) {
  int r = blockIdx.x * 8 + (threadIdx.x >> 5);  // r over B*NP
  int lane = threadIdx.x & 31;
  int b = r / NP_, i = r % NP_;
  const float* row = simfinal + (size_t)b * NPP_ + (size_t)i * NP_;
  float m = -3.4e38f;
  int mi = 0x7fffffff;
  for (int j = lane; j < NP_; j += 32) {
    float v = row[j];
    if (v > m || (v == m && j < mi)) { m = v; mi = j; }
  }
  for (int o = 16; o; o >>= 1) {
    float om = __shfl_xor(m, o, 32);
    int oi = __shfl_xor(mi, o, 32);
    if (om > m || (om == m && oi < mi)) { m = om; mi = oi; }
  }
  float Z = 0.f, dot = 0.f;
  for (int j = lane; j < NP_; j += 32) {
    float v = row[j];
    float e = __expf(v - m);
    Z += e;
    dot += e * v;
  }
  for (int o = 16; o; o >>= 1) { Z += __shfl_xor(Z, o, 32); dot += __shfl_xor(dot, o, 32); }
  if (!lane) {
    corr[r] = mi;
    float logZ = logf(Z);
    float logp = row[i] - m - logZ;  // diagonal log-softmax
    atomicAdd(&lossacc[2], -logp * matchg[r]);
    if (isFirst) {
      float ent = dot / Z - m - logZ;  // sum p log p
      float d = ssig[r] - ent;
      atomicAdd(&lossacc[3], d * d);
    }
  }
}

__global__ void col_softmax_kernel(const float* __restrict__ simfinal,
                                   const float* __restrict__ tsig, float* __restrict__ lossacc) {
  int r = blockIdx.x * 8 + (threadIdx.x >> 5);
  int lane = threadIdx.x & 31;
  int b = r / NP_, j = r % NP_;
  const float* base = simfinal + (size_t)b * NPP_ + j;
  float m = -3.4e38f;
  for (int k = lane; k < NP_; k += 32) m = fmaxf(m, base[(size_t)k * NP_]);
  for (int o = 16; o; o >>= 1) m = fmaxf(m, __shfl_xor(m, o, 32));
  float Z = 0.f, dot = 0.f;
  for (int k = lane; k < NP_; k += 32) {
    float v = base[(size_t)k * NP_];
    float e = __expf(v - m);
    Z += e;
    dot += e * v;
  }
  for (int o = 16; o; o >>= 1) { Z += __shfl_xor(Z, o, 32); dot += __shfl_xor(dot, o, 32); }
  if (!lane) {
    float ent = dot / Z - m - logf(Z);
    float d = tsig[r] - ent;
    atomicAdd(&lossacc[4], d * d);
  }
}

// ======================= wfc output: BCE + sigmoid =========================
__global__ void wfc_out_kernel(const float* __restrict__ whs, const float* __restrict__ scl,
                               const float* __restrict__ shf, const float* __restrict__ Wt,
                               const float* __restrict__ bias, const int* __restrict__ corr,
                               float* __restrict__ wsig, float* __restrict__ lossacc) {
  int t = blockIdx.x * 256 + threadIdx.x;
  if (t >= B_ * NP_) return;
  int b = t / NP_, k = t % NP_;
  float s = bias[0];
#pragma unroll
  for (int c = 0; c < 32; ++c) {
    float h = whs[((size_t)(b * 32 + c)) * NP_ + k];
    s += Wt[c] * fmaxf(h * scl[c] + shf[c], 0.f);
  }
  float lab = (corr[t] == k) ? 1.f : 0.f;
  float bce = fmaxf(s, 0.f) - s * lab + log1pf(__expf(-fabsf(s)));
  atomicAdd(&lossacc[5], bce);
  wsig[t] = 1.f / (1.f + __expf(-s));
}

// ============== median threshold + normalize (one block per b) =============
__global__ void median_norm_kernel(const float* __restrict__ wsig, float* __restrict__ wnorm) {
  __shared__ float wv[NP_];
  __shared__ float med, stot;
  int b = blockIdx.x;
  int tid = threadIdx.x;
  for (int k = tid; k < NP_; k += 256) wv[k] = wsig[b * NP_ + k];
  if (tid == 0) stot = 0.f;
  __syncthreads();
  for (int k = tid; k < NP_; k += 256) {
    float x = wv[k];
    int rank = 0;
    for (int j = 0; j < NP_; ++j) {
      float y = wv[j];
      rank += (y < x) || (y == x && j < k);
    }
    if (rank == (NP_ - 1) / 2) med = x;  // sorted[(NP-1)//2]
  }
  __syncthreads();
  float part = 0.f;
  for (int k = tid; k < NP_; k += 256) {
    float x = wv[k];
    x = (x >= med) ? x : 0.f;
    wv[k] = x;
    part += x;
  }
  atomicAdd(&stot, part);
  __syncthreads();
  float inv = 1.f / (stot + 1e-8f);
  for (int k = tid; k < NP_; k += 256) wnorm[b * NP_ + k] = wv[k] * inv;
}

// ================= weighted Kabsch (3x3 Jacobi SVD) + update ===============
__global__ void svd_update_kernel(const int* __restrict__ corr, const float* __restrict__ tgtp,
                                  const float* __restrict__ wnorm, float* __restrict__ srccur,
                                  float* __restrict__ Rcur, float* __restrict__ tcur) {
  int b = threadIdx.x;
  if (b >= B_) return;
  const float* sc = srccur + (size_t)b * 3 * NP_;
  const float* tp = tgtp + (size_t)b * 3 * NP_;
  const int* cr = corr + (size_t)b * NP_;
  const float* w = wnorm + (size_t)b * NP_;
  float ms[3] = {0, 0, 0}, mc[3] = {0, 0, 0}, wss[3] = {0, 0, 0}, wsc[3] = {0, 0, 0};
  for (int n = 0; n < NP_; ++n) {
    int ci = cr[n];
    float wv = w[n];
#pragma unroll
    for (int d = 0; d < 3; ++d) {
      float s = sc[d * NP_ + n];
      float c = tp[d * NP_ + ci];
      ms[d] += s; mc[d] += c;
      wss[d] += wv * s; wsc[d] += wv * c;
    }
  }
#pragma unroll
  for (int d = 0; d < 3; ++d) { ms[d] *= (1.f / NP_); mc[d] *= (1.f / NP_); }
  float H[3][3] = {{0, 0, 0}, {0, 0, 0}, {0, 0, 0}};
  for (int n = 0; n < NP_; ++n) {
    int ci = cr[n];
    float wv = w[n];
    float s0 = sc[0 * NP_ + n] - ms[0], s1 = sc[1 * NP_ + n] - ms[1], s2 = sc[2 * NP_ + n] - ms[2];
    float c0 = tp[0 * NP_ + ci] - mc[0], c1 = tp[1 * NP_ + ci] - mc[1], c2 = tp[2 * NP_ + ci] - mc[2];
    H[0][0] += wv * s0 * c0; H[0][1] += wv * s0 * c1; H[0][2] += wv * s0 * c2;
    H[1][0] += wv * s1 * c0; H[1][1] += wv * s1 * c1; H[1][2] += wv * s1 * c2;
    H[2][0] += wv * s2 * c0; H[2][1] += wv * s2 * c1; H[2][2] += wv * s2 * c2;
  }
  float A[3][3];
#pragma unroll
  for (int i = 0; i < 3; ++i)
#pragma unroll
    for (int j = 0; j < 3; ++j)
      A[i][j] = H[0][i] * H[0][j] + H[1][i] * H[1][j] + H[2][i] * H[2][j];
  float V[3][3] = {{1, 0, 0}, {0, 1, 0}, {0, 0, 1}};
  for (int sweep = 0; sweep < 12; ++sweep) {
    for (int pi = 0; pi < 3; ++pi) {
      int p = (pi == 2) ? 1 : 0;
      int q = (pi == 0) ? 1 : 2;
      float apq = A[p][q];
      if (fabsf(apq) < 1e-18f) continue;
      float theta = (A[q][q] - A[p][p]) / (2.f * apq);
      float tt = copysignf(1.f, theta) / (fabsf(theta) + sqrtf(theta * theta + 1.f));
      float c = rsqrtf(tt * tt + 1.f);
      float s = tt * c;
      for (int k = 0; k < 3; ++k) { float a0 = A[k][p], a1 = A[k][q]; A[k][p] = c * a0 - s * a1; A[k][q] = s * a0 + c * a1; }
      for (int k = 0; k < 3; ++k) { float a0 = A[p][k], a1 = A[q][k]; A[p][k] = c * a0 - s * a1; A[q][k] = s * a0 + c * a1; }
      for (int k = 0; k < 3; ++k) { float v0 = V[k][p], v1 = V[k][q]; V[k][p] = c * v0 - s * v1; V[k][q] = s * v0 + c * v1; }
    }
  }
  float lam[3] = {A[0][0], A[1][1], A[2][2]};
  for (int i = 0; i < 2; ++i)
    for (int j = i + 1; j < 3; ++j)
      if (lam[j] > lam[i]) {
        float tl = lam[i]; lam[i] = lam[j]; lam[j] = tl;
        for (int k = 0; k < 3; ++k) { float tv = V[k][i]; V[k][i] = V[k][j]; V[k][j] = tv; }
      }
  float sg[3], U[3][3];
#pragma unroll
  for (int i = 0; i < 3; ++i) sg[i] = sqrtf(fmaxf(lam[i], 0.f));
  for (int i = 0; i < 3; ++i) {
    if (sg[i] > 1e-12f) {
      float inv = 1.f / sg[i];
      for (int r = 0; r < 3; ++r)
        U[r][i] = (H[r][0] * V[0][i] + H[r][1] * V[1][i] + H[r][2] * V[2][i]) * inv;
    } else {
      int a0 = (i + 1) % 3, a1 = (i + 2) % 3;
      U[0][i] = U[1][a0] * U[2][a1] - U[2][a0] * U[1][a1];
      U[1][i] = U[2][a0] * U[0][a1] - U[0][a0] * U[2][a1];
      U[2][i] = U[0][a0] * U[1][a1] - U[1][a0] * U[0][a1];
      float nrm = rsqrtf(fmaxf(U[0][i] * U[0][i] + U[1][i] * U[1][i] + U[2][i] * U[2][i], 1e-30f));
      for (int r = 0; r < 3; ++r) U[r][i] *= nrm;
    }
  }
  float R0[3][3];
  for (int i = 0; i < 3; ++i)
    for (int k = 0; k < 3; ++k)
      R0[i][k] = V[i][0] * U[k][0] + V[i][1] * U[k][1] + V[i][2] * U[k][2];
  float det = R0[0][0] * (R0[1][1] * R0[2][2] - R0[1][2] * R0[2][1]) -
              R0[0][1] * (R0[1][0] * R0[2][2] - R0[1][2] * R0[2][0]) +
              R0[0][2] * (R0[1][0] * R0[2][1] - R0[1][1] * R0[2][0]);
  if (det < 0.f) {
    for (int k = 0; k < 3; ++k) V[k][2] = -V[k][2];
    for (int i = 0; i < 3; ++i)
      for (int k = 0; k < 3; ++k)
        R0[i][k] = V[i][0] * U[k][0] + V[i][1] * U[k][1] + V[i][2] * U[k][2];
  }
  float tr[3];
#pragma unroll
  for (int i = 0; i < 3; ++i)
    tr[i] = wsc[i] - (R0[i][0] * wss[0] + R0[i][1] * wss[1] + R0[i][2] * wss[2]);
  float* scw = srccur + (size_t)b * 3 * NP_;
  for (int n = 0; n < NP_; ++n) {
    float x0 = scw[0 * NP_ + n], x1 = scw[1 * NP_ + n], x2 = scw[2 * NP_ + n];
    scw[0 * NP_ + n] = R0[0][0] * x0 + R0[0][1] * x1 + R0[0][2] * x2 + tr[0];
    scw[1 * NP_ + n] = R0[1][0] * x0 + R0[1][1] * x1 + R0[1][2] * x2 + tr[1];
    scw[2 * NP_ + n] = R0[2][0] * x0 + R0[2][1] * x1 + R0[2][2] * x2 + tr[2];
  }
  float* Rb = Rcur + b * 9;
  float* tb = tcur + b * 3;
  float Rn[3][3], tn[3];
  for (int i = 0; i < 3; ++i) {
    for (int j = 0; j < 3; ++j)
      Rn[i][j] = R0[i][0] * Rb[0 * 3 + j] + R0[i][1] * Rb[1 * 3 + j] + R0[i][2] * Rb[2 * 3 + j];
    tn[i] = R0[i][0] * tb[0] + R0[i][1] * tb[1] + R0[i][2] * tb[2] + tr[i];
  }
  for (int i = 0; i < 3; ++i) {
    for (int j = 0; j < 3; ++j) Rb[i * 3 + j] = Rn[i][j];
    tb[i] = tn[i];
  }
}

__global__ void finalize_out_kernel(const float* __restrict__ Rcur, const float* __restrict__ tcur,
                                    const float* __restrict__ lossacc, float* __restrict__ out) {
  int t = threadIdx.x;
  if (t < 72) out[t] = Rcur[t];
  else if (t < 96) out[t] = tcur[t - 72];
  else if (t == 96) {
    float ms = lossacc[1];
    float loss = lossacc[2] / fmaxf(ms, 1e-12f) +
                 (lossacc[3] + lossacc[4] + lossacc[5]) / (float)(B_ * NP_);
    out[96] = loss;
  }
}

// ============================= host launcher ===============================
extern "C" void kernel_launch(void* const* d_in, const int* in_sizes, int n_in,
                              void* d_out, int out_size, void* d_ws, size_t ws_size,
                              hipStream_t stream) {
  (void)in_sizes; (void)n_in; (void)out_size; (void)ws_size;
  float* Wk = (float*)d_ws;
  const float* src = (const float*)d_in[IN_SRC];
  const float* tgt = (const float*)d_in[IN_TGT];
  const float* Rgt = (const float*)d_in[IN_RGT];
  const float* tgt_t = (const float*)d_in[IN_TGT_T];
  const int* srcidx = (const int*)d_in[IN_SRCIDX];
  auto F = [&](int i) { return (const float*)d_in[i]; };

  float* loss = Wk + O_LOSS;
  float* sum0 = Wk + O_SUM0; float* ssq0 = Wk + O_SSQ0;
  float* scl0 = Wk + O_SCL0; float* shf0 = Wk + O_SHF0;
  float* sum1 = Wk + O_SUM1; float* ssq1 = Wk + O_SSQ1;
  float* scl1 = Wk + O_SCL1; float* shf1 = Wk + O_SHF1;
  int* minidx = (int*)(Wk + O_MINIDX);
  int* corr = (int*)(Wk + O_CORR);

  fill_kernel<<<1, 256, 0, stream>>>(loss, 0.f, 8);
  corr_kernel<<<dim3(N_ / 256, B_), 256, 0, stream>>>(src, tgt, Rgt, tgt_t, minidx, Wk + O_MATCHF);

  const float cntN = (float)((size_t)B_ * N_);
  auto emb_chain = [&](const float* x, float* outEmb) {
    fill_kernel<<<1, 256, 0, stream>>>(sum0, 0.f, 256);
    convbn_kernel<4, 3, 64, false, false, true><<<dim3(16, 4, B_), 256, 0, stream>>>(
        x, F(IN_EMB0_W), nullptr, nullptr, nullptr, Wk + O_HB, sum0, ssq0, N_);
    bn_finalize_kernel<<<1, 128, 0, stream>>>(sum0, ssq0, F(IN_EMB0_G), F(IN_EMB0_B), scl0, shf0, 64, cntN);
    fill_kernel<<<1, 256, 0, stream>>>(sum1, 0.f, 256);
    convbn_kernel<64, 64, 128, true, false, true><<<dim3(16, 8, B_), 256, 0, stream>>>(
        Wk + O_HB, F(IN_EMB1_W), nullptr, scl0, shf0, Wk + O_HA, sum1, ssq1, N_);
    bn_finalize_kernel<<<1, 128, 0, stream>>>(sum1, ssq1, F(IN_EMB1_G), F(IN_EMB1_B), scl1, shf1, 128, cntN);
    convbn_kernel<128, 128, 64, true, true, false><<<dim3(16, 4, B_), 256, 0, stream>>>(
        Wk + O_HA, F(IN_EMB2_W), F(IN_EMB2_BIAS), scl1, shf1, outEmb, nullptr, nullptr, N_);
  };
  emb_chain(tgt, Wk + O_TEMB);
  emb_chain(src, Wk + O_SEMB);

  auto sig_chain = [&](const float* emb, float* outSig) {
    fill_kernel<<<1, 256, 0, stream>>>(sum0, 0.f, 256);
    convbn_kernel<64, 64, 64, false, false, true><<<dim3(16, 4, B_), 256, 0, stream>>>(
        emb, F(IN_SIG0_W), nullptr, nullptr, nullptr, Wk + O_HB, sum0, ssq0, N_);
    bn_finalize_kernel<<<1, 128, 0, stream>>>(sum0, ssq0, F(IN_SIG0_G), F(IN_SIG0_B), scl0, shf0, 64, cntN);
    fill_kernel<<<1, 256, 0, stream>>>(sum1, 0.f, 256);
    convbn_kernel<64, 64, 32, true, false, true><<<dim3(16, 2, B_), 256, 0, stream>>>(
        Wk + O_HB, F(IN_SIG1_W), nullptr, scl0, shf0, Wk + O_HA, sum1, ssq1, N_);
    bn_finalize_kernel<<<1, 128, 0, stream>>>(sum1, ssq1, F(IN_SIG1_G), F(IN_SIG1_B), scl1, shf1, 32, cntN);
    conv_last1_kernel<<<(B_ * N_ + 255) / 256, 256, 0, stream>>>(
        Wk + O_HA, scl1, shf1, F(IN_SIG2_W), F(IN_SIG2_BIAS), outSig, N_);
  };
  sig_chain(Wk + O_TEMB, Wk + O_TSIGF);
  sig_chain(Wk + O_SEMB, Wk + O_SSIGF);

  gather_kernel<<<(B_ * NP_ + 255) / 256, 256, 0, stream>>>(
      src, tgt, srcidx, minidx, Wk + O_MATCHF, Wk + O_SEMB, Wk + O_TEMB, Wk + O_SSIGF, Wk + O_TSIGF,
      Wk + O_SRCP, Wk + O_TGTP, Wk + O_SEMBG, Wk + O_TEMBG, Wk + O_SSIGG, Wk + O_TSIGG,
      Wk + O_MATCHG, Wk + O_SRCCUR, loss);
  init_Rt_kernel<<<1, 128, 0, stream>>>(Wk + O_RCUR, Wk + O_TCUR);

  const float cntPP = (float)((size_t)B_ * NPP_);
  for (int it = 0; it < 3; ++it) {
    const float* s1b = F(IN_SIM1 + 5 * it + 0);
    const float* s1g = F(IN_SIM1 + 5 * it + 1);
    const float* w1a = F(IN_SIM1 + 5 * it + 2);
    const float* b1b = F(IN_SIM1 + 5 * it + 3);
    const float* w1b = F(IN_SIM1 + 5 * it + 4);
    const float* s2b = F(IN_SIM2 + 5 * it + 0);
    const float* s2g = F(IN_SIM2 + 5 * it + 1);
    const float* w2a = F(IN_SIM2 + 5 * it + 2);
    const float* b2b = F(IN_SIM2 + 5 * it + 3);
    const float* w2b = F(IN_SIM2 + 5 * it + 4);
    const float* wfb = F(IN_WFC + 5 * it + 0);
    const float* wfg = F(IN_WFC + 5 * it + 1);
    const float* wfw = F(IN_WFC + 5 * it + 2);
    const float* wf1b = F(IN_WFC + 5 * it + 3);
    const float* wf1w = F(IN_WFC + 5 * it + 4);

    fill_kernel<<<1, 256, 0, stream>>>(sum0, 0.f, 256);
    sim_h1_stats_kernel<<<dim3(NPBLK, B_), 256, 0, stream>>>(
        w1a, Wk + O_SEMBG, Wk + O_TEMBG, Wk + O_SRCCUR, Wk + O_TGTP, sum0, ssq0);
    bn_finalize_kernel<<<1, 128, 0, stream>>>(sum0, ssq0, s1g, s1b, scl0, shf0, 64, cntPP);

    fill_kernel<<<1, 256, 0, stream>>>(sum1, 0.f, 256);
    sim_feat_kernel<<<dim3(NPBLK, B_), 256, 0, stream>>>(
        w1a, scl0, shf0, w1b, b1b, w2a, Wk + O_SEMBG, Wk + O_TEMBG, Wk + O_SRCCUR, Wk + O_TGTP,
        Wk + O_SIMFEA, sum1, ssq1);
    bn_finalize_kernel<<<1, 128, 0, stream>>>(sum1, ssq1, s2g, s2b, scl1, shf1, 32, cntPP);

    rowmax_kernel<<<(B_ * 64 * NP_) / 8, 256, 0, stream>>>(Wk + O_SIMFEA, Wk + O_W64);
    sim_final_kernel<<<dim3(NPBLK, B_), 256, 0, stream>>>(
        Wk + O_SIMFEA, w2a, scl1, shf1, w2b, b2b, Wk + O_SIMFIN);

    row_softmax_kernel<<<(B_ * NP_) / 8, 256, 0, stream>>>(
        Wk + O_SIMFIN, Wk + O_MATCHG, Wk + O_SSIGG, corr, loss, it == 0 ? 1 : 0);
    if (it == 0)
      col_softmax_kernel<<<(B_ * NP_) / 8, 256, 0, stream>>>(Wk + O_SIMFIN, Wk + O_TSIGG, loss);

    fill_kernel<<<1, 256, 0, stream>>>(sum0, 0.f, 256);
    convbn_kernel<64, 64, 32, false, false, true><<<dim3(2, 2, B_), 256, 0, stream>>>(
        Wk + O_W64, wfw, nullptr, nullptr, nullptr, Wk + O_WHS, sum0, ssq0, NP_);
    bn_finalize_kernel<<<1, 128, 0, stream>>>(sum0, ssq0, wfg, wfb, scl0, shf0, 32, (float)(B_ * NP_));
    wfc_out_kernel<<<(B_ * NP_ + 255) / 256, 256, 0, stream>>>(
        Wk + O_WHS, scl0, shf0, wf1w, wf1b, corr, Wk + O_WSIG, loss);
    median_norm_kernel<<<B_, 256, 0, stream>>>(Wk + O_WSIG, Wk + O_WNORM);
    svd_update_kernel<<<1, 64, 0, stream>>>(corr, Wk + O_TGTP, Wk + O_WNORM,
                                            Wk + O_SRCCUR, Wk + O_RCUR, Wk + O_TCUR);
  }
  finalize_out_kernel<<<1, 128, 0, stream>>>(Wk + O_RCUR, Wk + O_TCUR, loss, (float*)d_out);
}